// exam_fc_54735063220554
// MI455X (gfx1250) — compile-verified
//
#include <hip/hip_runtime.h>
#include <hip/hip_bf16.h>

// ---------------------------------------------------------------------------
// Fused 19-head Linear+ReLU == one GEMM: y = relu(x[65536,1024] @ W[1024,84] + b)
//
// Memory-bound (AI ~40 FLOP/B, bound ~12us @ 23.3 TB/s), so: exact fp32 math on
// V_WMMA_F32_16X16X4_F32, x streamed once with nontemporal b128 loads, W packed
// once into WMMA B-fragment layout in d_ws so every B fragment is one b128 load.
// ---------------------------------------------------------------------------

typedef float v2f __attribute__((ext_vector_type(2)));
typedef float v4f __attribute__((ext_vector_type(4)));
typedef float v8f __attribute__((ext_vector_type(8)));

#define FEAT   1024
#define NOUT   84      // sum of the 19 head dims
#define NPAD   96      // padded to 6 tiles of 16
#define NTILES 6
#define K8S    (FEAT / 8)   // 128 macro-steps of K=8 (2 WMMAs each)
#define BATCH  65536

// Pack W[1024,84] (row-major, K x N) into B-fragment layout, zero-padding
// columns 84..95.  For macro-step k8 and lane-half h, lane with n = l&15 reads
// v4f Wp[(k8*2+h)*96 + n] = { W[k8*8+4h+j][n] : j=0..3 }.
// (Within a WMMA, lanes<16 supply K-pos {0,1}, lanes>=16 supply {2,3}; we use
// the logical K permutation {k,k+1,k+4,k+5} / {k+2,k+3,k+6,k+7} consistently
// for A and B, which leaves the K-reduction result unchanged.)
__global__ void pack_w_kernel(const float* __restrict__ W, float* __restrict__ Wp) {
    int t = blockIdx.x * blockDim.x + threadIdx.x;
    if (t >= K8S * 2 * NPAD) return;
    int n  = t % NPAD;
    int h  = (t / NPAD) & 1;
    int k8 = t / (2 * NPAD);
    int kb = k8 * 8 + 4 * h;
    v4f v;
#pragma unroll
    for (int j = 0; j < 4; ++j)
        v[j] = (n < NOUT) ? W[(size_t)(kb + j) * NOUT + n] : 0.0f;
    reinterpret_cast<v4f*>(Wp)[t] = v;
}

template <bool PACKED>
__global__ __launch_bounds__(256)
void fused_heads_gemm(const float* __restrict__ x,
                      const float* __restrict__ Bsrc,   // Wp if PACKED else W
                      const float* __restrict__ bias,
                      float* __restrict__ out) {
    const int lane = threadIdx.x & 31;
    const int wave = threadIdx.x >> 5;
    const int h    = lane >> 4;    // lane half: supplies K-pos {0,1} or {2,3}
    const int nl   = lane & 15;
    const int row0 = (blockIdx.x * 8 + wave) * 16;      // 16-row stripe per wave

    const float* aptr = x + (size_t)(row0 + nl) * FEAT + 4 * h;

    v8f acc[NTILES] = {};

    for (int k8 = 0; k8 < K8S; ++k8) {
        // A fragment: 4 contiguous floats per lane (x streamed once, no reuse).
        v4f a = __builtin_nontemporal_load(
            reinterpret_cast<const v4f*>(aptr + k8 * 8));
        v2f a01; a01[0] = a[0]; a01[1] = a[1];
        v2f a23; a23[0] = a[2]; a23[1] = a[3];
#pragma unroll
        for (int t = 0; t < NTILES; ++t) {
            v2f b01, b23;
            if (PACKED) {
                const v4f bv = reinterpret_cast<const v4f*>(Bsrc)
                    [(size_t)(k8 * 2 + h) * NPAD + t * 16 + nl];
                b01[0] = bv[0]; b01[1] = bv[1];
                b23[0] = bv[2]; b23[1] = bv[3];
            } else {
                // Fallback: strided loads straight from W with zero guard.
                int  n  = t * 16 + nl;
                int  kb = k8 * 8 + 4 * h;
                bool ok = (n < NOUT);
                b01[0] = ok ? Bsrc[(size_t)(kb + 0) * NOUT + n] : 0.0f;
                b01[1] = ok ? Bsrc[(size_t)(kb + 1) * NOUT + n] : 0.0f;
                b23[0] = ok ? Bsrc[(size_t)(kb + 2) * NOUT + n] : 0.0f;
                b23[1] = ok ? Bsrc[(size_t)(kb + 3) * NOUT + n] : 0.0f;
            }
            // D = A*B + C, exact fp32.  8 args:
            // (neg_a, A, neg_b, B, c_mod, C, reuse_a, reuse_b)
            acc[t] = __builtin_amdgcn_wmma_f32_16x16x4_f32(
                false, a01, false, b01, (short)0, acc[t], false, false);
            acc[t] = __builtin_amdgcn_wmma_f32_16x16x4_f32(
                false, a23, false, b23, (short)0, acc[t], false, false);
        }
    }

    // Epilogue: bias + ReLU + store (only valid 84 columns). C/D layout:
    // lane -> (M = r + 8h, N = l&15) for VGPR r.  Divergence here is fine,
    // all WMMAs are already done.
#pragma unroll
    for (int t = 0; t < NTILES; ++t) {
        int n = t * 16 + nl;
        if (n < NOUT) {
            float bv = bias[n];
#pragma unroll
            for (int r = 0; r < 8; ++r) {
                int   m = row0 + r + 8 * h;
                float v = acc[t][r] + bv;
                v = v > 0.0f ? v : 0.0f;
                __builtin_nontemporal_store(v, out + (size_t)m * NOUT + n);
            }
        }
    }
}

extern "C" void kernel_launch(void* const* d_in, const int* in_sizes, int n_in,
                              void* d_out, int out_size, void* d_ws, size_t ws_size,
                              hipStream_t stream) {
    (void)in_sizes; (void)n_in; (void)out_size;
    const float* x    = (const float*)d_in[0];
    const float* W    = (const float*)d_in[1];
    const float* bias = (const float*)d_in[2];
    float*       out  = (float*)d_out;

    const size_t pack_bytes = (size_t)K8S * 2 * NPAD * 4 * sizeof(float); // 384 KB
    const int    nblocks    = BATCH / 128;   // 8 waves/block * 16 rows/wave

    if (d_ws && ws_size >= pack_bytes) {
        float* Wp = (float*)d_ws;
        int pack_threads = K8S * 2 * NPAD;   // 24576
        pack_w_kernel<<<(pack_threads + 255) / 256, 256, 0, stream>>>(W, Wp);
        fused_heads_gemm<true><<<nblocks, 256, 0, stream>>>(x, Wp, bias, out);
    } else {
        fused_heads_gemm<false><<<nblocks, 256, 0, stream>>>(x, W, bias, out);
    }
}